// LightRNNDecoder_33054068310800
// MI455X (gfx1250) — compile-verified
//
#include <hip/hip_runtime.h>
#include <hip/hip_bf16.h>

// ---------------------------------------------------------------------------
// LightRNN decoder loss for MI455X (gfx1250), wave32, WMMA f32 16x16x4.
//
// Strategy: bucket tokens by row id so each 768x256 col_weight slab is read
// once (197 MB instead of 805 MB -> ~9 us at 23.3 TB/s HBM). GEMMs use
// V_WMMA_F32_16X16X4_F32 tiles (exact fp32 numerics, compute is not the
// bottleneck). Per-token CE values are order-independent; final reduction is
// a fixed-shape tree -> deterministic output.
// ---------------------------------------------------------------------------

#define N_EMBD 768
#define TBL    256
#define LDA    772   // padded LDS stride (772 % 64 == 4 -> conflict-free rows)

typedef __attribute__((ext_vector_type(2))) float v2f;
typedef __attribute__((ext_vector_type(8))) float v8f;

// ---------------------------------------------------------------------------
// Shared GEMM core: one workgroup (256 thr = 8 waves) computes a 16x256 logits
// tile: ldsL[16][256] = ldsA[16][768(pad LDA)] @ Bmat[768][256] + bias[256].
// Each wave owns two 16x16 C tiles (8 waves x 2 iters = 256 columns).
// WMMA f32 16x16x4 operand layout per CDNA5 ISA 7.12.2:
//   A (16x4, 2 VGPR): lanes 0-15 -> M=lane, K={k0,k0+1}; lanes 16-31 -> K={k0+2,k0+3}
//   B (4x16, 2 VGPR): N = lane&15, same K split
//   C/D (8 VGPR): M = j + 8*(lane>>4), N = lane&15
// Must be called by all 256 threads (EXEC all ones).
// ---------------------------------------------------------------------------
__device__ __forceinline__ void compute_tile_logits(
    const float* __restrict__ ldsA,   // [16][LDA] in LDS
    float* __restrict__ ldsL,         // [16][256] in LDS
    const float* __restrict__ Bmat,   // [768][256] k-major in global
    const float* __restrict__ bias)   // [256]
{
  const int tid   = (int)threadIdx.x;
  const int wave  = tid >> 5;
  const int lane  = tid & 31;
  const int m     = lane & 15;
  const int khalf = (lane >> 4) << 1;   // 0 or 2
  const int mbase = (lane >> 4) << 3;   // 0 or 8 (C-tile row base)

  for (int iter = 0; iter < 2; ++iter) {
    const int n0 = (wave + (iter << 3)) << 4;   // column tile base
    const int n  = n0 + (lane & 15);
    v8f c = {};
    #pragma unroll 4
    for (int k0 = 0; k0 < N_EMBD; k0 += 4) {
      const int kb = k0 + khalf;
      v2f a = *reinterpret_cast<const v2f*>(ldsA + m * LDA + kb);
      v2f b;
      b.x = Bmat[(size_t)kb * TBL + n];
      b.y = Bmat[(size_t)(kb + 1) * TBL + n];
      c = __builtin_amdgcn_wmma_f32_16x16x4_f32(
          /*neg_a=*/false, a, /*neg_b=*/false, b,
          /*c_mod=*/(short)0, c, /*reuse_a=*/false, /*reuse_b=*/false);
    }
    const float bn = bias[n];
    #pragma unroll
    for (int j = 0; j < 8; ++j) {
      ldsL[(mbase + j) * TBL + n] = c[j] + bn;
    }
  }
}

// Per-token log-sum-exp CE over a 16x256 LDS logits tile (threads 0..nvalid-1).
__device__ __forceinline__ float ce_from_lds_row(const float* row, int label) {
  float mx = -3.402823466e+38f;
  for (int j = 0; j < TBL; ++j) mx = fmaxf(mx, row[j]);
  float s = 0.0f;
  for (int j = 0; j < TBL; ++j) s += __expf(row[j] - mx);
  return (__logf(s) + mx) - row[label];
}

// ---------------------------------------------------------------------------
// Prep kernels
// ---------------------------------------------------------------------------
__global__ void k_init(int* counts, int* cursor, float* out) {
  const int t = (int)threadIdx.x;
  counts[t] = 0;
  cursor[t] = 0;
  if (t == 0) out[0] = 0.0f;
}

__global__ void k_ids_hist(const int* __restrict__ target_ids, int n,
                           int* __restrict__ row_ids, int* __restrict__ col_ids,
                           int* __restrict__ counts) {
  const int i = (int)(blockIdx.x * blockDim.x + threadIdx.x);
  if (i >= n) return;
  const int v = target_ids[i];
  const int r = (v >> 8) & 255;
  row_ids[i] = r;
  col_ids[i] = v & 255;
  atomicAdd(&counts[r], 1);
}

__global__ void k_scan(const int* __restrict__ counts, int* __restrict__ offsets) {
  __shared__ int tmp[TBL];
  const int t = (int)threadIdx.x;
  const int v0 = counts[t];
  tmp[t] = v0;
  __syncthreads();
  for (int d = 1; d < TBL; d <<= 1) {
    const int v = (t >= d) ? tmp[t - d] : 0;
    __syncthreads();
    tmp[t] += v;
    __syncthreads();
  }
  offsets[t] = tmp[t] - v0;   // exclusive prefix
}

__global__ void k_scatter(const int* __restrict__ row_ids, int n,
                          const int* __restrict__ offsets, int* __restrict__ cursor,
                          int* __restrict__ bucket) {
  const int i = (int)(blockIdx.x * blockDim.x + threadIdx.x);
  if (i >= n) return;
  const int r = row_ids[i];
  const int p = atomicAdd(&cursor[r], 1);
  bucket[offsets[r] + p] = i;
}

// row_W is (256 out, 768 in) row-major; make a k-major (768x256) copy so the
// GEMM B-reads are coalesced like col_weight's native layout.
__global__ void k_transpose(const float* __restrict__ rowW, float* __restrict__ rowWt) {
  const int idx = (int)(blockIdx.x * blockDim.x + threadIdx.x);  // 768*256
  const int k = idx >> 8;
  const int t = idx & 255;
  rowWt[(size_t)k * TBL + t] = rowW[(size_t)t * N_EMBD + k];
}

// ---------------------------------------------------------------------------
// Row head: 64 blocks x 16 contiguous tokens each.
// ---------------------------------------------------------------------------
__global__ void k_row(const float* __restrict__ hs, const int* __restrict__ row_ids,
                      const float* __restrict__ rowWt, const float* __restrict__ row_b,
                      float* __restrict__ ce_row, int n_tok) {
  __shared__ float ldsA[16 * LDA];
  __shared__ float ldsL[16 * TBL];
  const int tok0 = (int)blockIdx.x * 16;

  for (int mrow = 0; mrow < 16; ++mrow) {
    const int tok = tok0 + mrow;
    for (int k = (int)threadIdx.x; k < N_EMBD; k += 256)
      ldsA[mrow * LDA + k] = (tok < n_tok) ? hs[(size_t)tok * N_EMBD + k] : 0.0f;
  }
  __syncthreads();

  compute_tile_logits(ldsA, ldsL, rowWt, row_b);
  __syncthreads();

  const int nvalid = min(16, n_tok - tok0);
  if ((int)threadIdx.x < nvalid) {
    const int tok = tok0 + (int)threadIdx.x;
    ce_row[tok] = ce_from_lds_row(ldsL + (int)threadIdx.x * TBL, row_ids[tok]);
  }
}

// ---------------------------------------------------------------------------
// Column head: one block per row id r; processes its bucketed tokens in chunks
// of 16, streaming the 768x256 col_weight[r] slab through WMMA once per chunk.
// ---------------------------------------------------------------------------
__global__ void k_col(const float* __restrict__ hs, const int* __restrict__ col_ids,
                      const float* __restrict__ colW, const float* __restrict__ col_b,
                      const int* __restrict__ counts, const int* __restrict__ offsets,
                      const int* __restrict__ bucket, float* __restrict__ ce_col) {
  __shared__ float ldsA[16 * LDA];
  __shared__ float ldsL[16 * TBL];
  const int r   = (int)blockIdx.x;
  const int cnt = counts[r];
  if (cnt == 0) return;                      // uniform per block

  const int   off  = offsets[r];
  const float* Bmat = colW  + (size_t)r * N_EMBD * TBL;
  const float* bias = col_b + (size_t)r * TBL;

  for (int base = 0; base < cnt; base += 16) {
    for (int mrow = 0; mrow < 16; ++mrow) {
      const int bi = base + mrow;            // uniform across block
      if (bi < cnt) {
        const int tok = bucket[off + bi];
        for (int k = (int)threadIdx.x; k < N_EMBD; k += 256)
          ldsA[mrow * LDA + k] = hs[(size_t)tok * N_EMBD + k];
      } else {
        for (int k = (int)threadIdx.x; k < N_EMBD; k += 256)
          ldsA[mrow * LDA + k] = 0.0f;
      }
    }
    __syncthreads();

    compute_tile_logits(ldsA, ldsL, Bmat, bias);
    __syncthreads();

    const int nvalid = min(16, cnt - base);
    if ((int)threadIdx.x < nvalid) {
      const int tok = bucket[off + base + (int)threadIdx.x];
      ce_col[tok] = ce_from_lds_row(ldsL + (int)threadIdx.x * TBL, col_ids[tok]);
    }
    __syncthreads();
  }
}

// Deterministic fixed-shape reduction: loss = mean(ce_row) + mean(ce_col).
__global__ void k_reduce(const float* __restrict__ ce_row, const float* __restrict__ ce_col,
                         float* __restrict__ out, int n, float inv_n) {
  __shared__ float sm[256];
  float s = 0.0f;
  for (int i = (int)threadIdx.x; i < n; i += 256) s += ce_row[i] + ce_col[i];
  sm[threadIdx.x] = s;
  __syncthreads();
  for (int d = 128; d > 0; d >>= 1) {
    if ((int)threadIdx.x < d) sm[threadIdx.x] += sm[threadIdx.x + d];
    __syncthreads();
  }
  if (threadIdx.x == 0) out[0] = sm[0] * inv_n;
}

// ---------------------------------------------------------------------------
extern "C" void kernel_launch(void* const* d_in, const int* in_sizes, int n_in,
                              void* d_out, int out_size, void* d_ws, size_t ws_size,
                              hipStream_t stream) {
  const float* hs         = (const float*)d_in[0];   // (B,T,768)
  const int*   target_ids = (const int*)  d_in[1];   // (B,T)
  const float* rowW       = (const float*)d_in[2];   // (256,768)
  const float* row_b      = (const float*)d_in[3];   // (256)
  const float* colW       = (const float*)d_in[4];   // (256,768,256)
  const float* col_b      = (const float*)d_in[5];   // (256,256)
  float* out = (float*)d_out;

  const int n_tok = in_sizes[1];                     // B*T = 1024
  const float inv_n = 1.0f / (float)n_tok;

  // Workspace carve-up (16B aligned slices).
  char* p = (char*)d_ws;
  auto carve = [&](size_t bytes) { char* q = p; p += (bytes + 15) & ~(size_t)15; return q; };
  int*   row_ids = (int*)  carve((size_t)n_tok * 4);
  int*   col_ids = (int*)  carve((size_t)n_tok * 4);
  int*   counts  = (int*)  carve(TBL * 4);
  int*   offsets = (int*)  carve(TBL * 4);
  int*   cursor  = (int*)  carve(TBL * 4);
  int*   bucket  = (int*)  carve((size_t)n_tok * 4);
  float* ce_row  = (float*)carve((size_t)n_tok * 4);
  float* ce_col  = (float*)carve((size_t)n_tok * 4);
  float* rowWt   = (float*)carve((size_t)N_EMBD * TBL * 4);
  (void)ws_size; (void)n_in; (void)out_size;

  const int nblk = (n_tok + 255) / 256;

  k_init     <<<1,                 256, 0, stream>>>(counts, cursor, out);
  k_ids_hist <<<nblk,              256, 0, stream>>>(target_ids, n_tok, row_ids, col_ids, counts);
  k_scan     <<<1,                 256, 0, stream>>>(counts, offsets);
  k_scatter  <<<nblk,              256, 0, stream>>>(row_ids, n_tok, offsets, cursor, bucket);
  k_transpose<<<(N_EMBD*TBL)/256,  256, 0, stream>>>(rowW, rowWt);
  k_row      <<<(n_tok + 15)/16,   256, 0, stream>>>(hs, row_ids, rowWt, row_b, ce_row, n_tok);
  k_col      <<<TBL,               256, 0, stream>>>(hs, col_ids, colW, col_b, counts, offsets, bucket, ce_col);
  k_reduce   <<<1,                 256, 0, stream>>>(ce_row, ce_col, out, n_tok, inv_n);
}